// GAT_34600256537462
// MI455X (gfx1250) — compile-verified
//
#include <hip/hip_runtime.h>
#include <math.h>
#include <stdint.h>

#define N_NODES 100000
#define N_EDGES 1600000
#define ETOT    (N_EDGES + N_NODES)
#define F_IN    64
#define HID     128
#define HEADS   8
#define NGRAPH  1000

typedef __attribute__((ext_vector_type(2))) float v2f;
typedef __attribute__((ext_vector_type(8))) float v8f;
typedef __attribute__((ext_vector_type(4))) unsigned int u32x4;
typedef __attribute__((ext_vector_type(8))) int i32x8;
typedef __attribute__((ext_vector_type(4))) int i32x4;

static inline int cdiv(long long a, long long b) { return (int)((a + b - 1) / b); }

__device__ __forceinline__ float elu_f(float x) { return x > 0.f ? x : (__expf(x) - 1.f); }
__device__ __forceinline__ float lrelu_f(float x) { return x > 0.f ? x : 0.2f * x; }

// ---------------------------------------------------------------------------
// TDM: DMA a 2D f32 tile [height x width] (row stride `stride` elements)
// from global memory into LDS at byte offset lds_off. D# packing per CDNA5
// ISA 8.3/8.4: group0 = {count, lds_addr, global_addr, type=2},
// group1 = {data_size=4B, tensor_dim0/1, tile_dim0/1/2, tensor_dim0_stride}.
// Groups 2/3 zero (tensor <= 2D). Tracked by TENSORcnt.
// Toolchain uses the 6-arg builtin: (u32x4, i32x8, i32x4, i32x4, i32x8, i32).
// ---------------------------------------------------------------------------
__device__ __forceinline__ void tdm_load_2d_f32(const float* gsrc, unsigned lds_off,
                                                unsigned width, unsigned height,
                                                unsigned stride)
{
    unsigned long long ga = (unsigned long long)(uintptr_t)gsrc;
    u32x4 g0;
    g0[0] = 1u;                                             // count=1, user descriptor
    g0[1] = lds_off;                                        // lds_addr (bytes)
    g0[2] = (unsigned)ga;                                   // global_addr[31:0]
    g0[3] = (unsigned)((ga >> 32) & 0x01FFFFFFu) | (2u << 30); // addr[56:32] | type=2
    i32x8 g1;
    g1[0] = (2 << 16);                                      // data_size: 2 -> 4 bytes
    g1[1] = (int)((width & 0xFFFFu) << 16);                 // tensor_dim0[15:0] @ bits63:48
    g1[2] = (int)((width >> 16) | ((height & 0xFFFFu) << 16)); // dim0[31:16], dim1[15:0]
    g1[3] = (int)((height >> 16) | (width << 16));          // dim1[31:16], tile_dim0
    g1[4] = (int)(height & 0xFFFFu);                        // tile_dim1 (tile_dim2 = 0)
    g1[5] = (int)stride;                                    // tensor_dim0_stride[31:0]
    g1[6] = 0;                                              // stride0[47:32], stride1[15:0]
    g1[7] = 0;                                              // stride1[47:16]
    i32x4 gz4 = {0, 0, 0, 0};
    i32x8 gz8 = {0, 0, 0, 0, 0, 0, 0, 0};
    __builtin_amdgcn_tensor_load_to_lds(g0, g1, gz4, gz4, gz8, 0);
}

// ---------------------------------------------------------------------------
// WMMA fp32 GEMM: D[M x 128] = A[M x K] @ B[K x 128] (+bias, optional ELU)
// Block = 256 threads = 8 waves = 4 row-tiles x 2 column-halves.
// B (K x 128) staged once per block into LDS by the Tensor Data Mover; the
// K-loop then feeds V_WMMA_F32_16X16X4_F32 from ds_load (B) + pipelined
// global loads (A). K templated -> full unroll, immediate-offset addressing.
// A-frag (ISA 7.12.2, 32-bit A 16x4): lane L -> row (L&15), k-off (L>>4)*2.
// C/D: vgpr j -> row (L>>4)*8 + j, col (L&15).
// ---------------------------------------------------------------------------
template <int K, int NT>
__global__ __launch_bounds__(256)
void wmma_gemm_k(const float* __restrict__ A, const float* __restrict__ B,
                 float* __restrict__ D, const float* __restrict__ bias,
                 int M, int applyElu)
{
    __shared__ float Bs[K * HID];                 // 32KB (K=64) / 64KB (K=128)

    const int wave = threadIdx.x >> 5;
    const int lane = threadIdx.x & 31;

    if (wave == 0) {                              // one TDM descriptor per block
        tdm_load_2d_f32(B, (unsigned)(uintptr_t)(void*)Bs, HID, K, HID);
        __builtin_amdgcn_s_wait_tensorcnt((short)0);
    }
    __syncthreads();

    const int tm     = blockIdx.x * 4 + (wave >> 1);
    const int tnBase = (wave & 1) * NT;
    const int r      = lane & 15;
    const int koff   = (lane >> 4) * 2;

    const int rowA = tm * 16 + r;
    const int rowC = rowA < M ? rowA : (M - 1);   // clamp: dup rows never stored
    const float2* __restrict__ ap = (const float2*)(A + (size_t)rowC * K);

    // prefetch next block's A rows (global_prefetch_b8)
    if (rowA + 64 < M) __builtin_prefetch(A + (size_t)(rowC + 64) * K, 0, 1);

    v8f acc[NT] = {};

#pragma unroll
    for (int k = 0; k < K; k += 4) {
        const float2 af = ap[(k + koff) >> 1];
        v2f a; a.x = af.x; a.y = af.y;
#pragma unroll
        for (int t = 0; t < NT; ++t) {
            const int col = (tnBase + t) * 16 + r;
            v2f b;
            b.x = Bs[(k + koff)     * HID + col];
            b.y = Bs[(k + koff + 1) * HID + col];
            acc[t] = __builtin_amdgcn_wmma_f32_16x16x4_f32(false, a, false, b,
                                                           (short)0, acc[t], false, false);
        }
    }

    const int rbase = tm * 16 + (lane >> 4) * 8;
#pragma unroll
    for (int t = 0; t < NT; ++t) {
        const int col = (tnBase + t) * 16 + r;
        const float bv = bias ? bias[col] : 0.f;
#pragma unroll
        for (int j = 0; j < 8; ++j) {
            const int row = rbase + j;
            if (row < M) {
                float v = acc[t][j] + bv;
                if (applyElu) v = elu_f(v);
                D[(size_t)row * HID + col] = v;
            }
        }
    }
}

// ---------------------------------------------------------------------------
// alpha_s/alpha_d: one wave per node, float4 loads, segmented shfl_xor
// reduction over groups of (C/4) lanes. a_src/a_dst flattened [H,C] match
// the channel layout of h exactly, so it's an elementwise product + segsum.
// ---------------------------------------------------------------------------
__global__ __launch_bounds__(256)
void gat_alpha(const float* __restrict__ h,
               const float* __restrict__ a_src, const float* __restrict__ a_dst,
               float* __restrict__ asn, float* __restrict__ adn,
               int H, int C, int n)
{
    const int wave = threadIdx.x >> 5;
    const int lane = threadIdx.x & 31;
    const int row  = blockIdx.x * 8 + wave;
    if (row >= n) return;

    const float4 hv = ((const float4*)(h + (size_t)row * HID))[lane];
    const float4 sv = ((const float4*)a_src)[lane];
    const float4 dv = ((const float4*)a_dst)[lane];

    float s = hv.x * sv.x + hv.y * sv.y + hv.z * sv.z + hv.w * sv.w;
    float d = hv.x * dv.x + hv.y * dv.y + hv.z * dv.z + hv.w * dv.w;

    const int width = C >> 2;                 // lanes per head: 4 (C=16) or 32 (C=128)
    for (int off = width >> 1; off > 0; off >>= 1) {
        s += __shfl_xor(s, off, 32);
        d += __shfl_xor(d, off, 32);
    }
    if ((lane & (width - 1)) == 0) {
        const int hd = lane / width;
        asn[row * H + hd] = s;
        adn[row * H + hd] = d;
    }
}

__global__ void fill_f32(float* __restrict__ p, float v, int n)
{
    int i = blockIdx.x * blockDim.x + threadIdx.x;
    if (i < n) p[i] = v;
}

// ---------------------------------------------------------------------------
// Edge pass 1: per-dst segment max of leaky_relu(alpha_s[src]+alpha_d[dst]).
// Float atomic max via monotonic int mapping (init = -inf).
// ---------------------------------------------------------------------------
__global__ void gat_edge_max(const int* __restrict__ esrc, const int* __restrict__ edst,
                             const float* __restrict__ asn, const float* __restrict__ adn,
                             float* __restrict__ mx, int hshift, int total)
{
    int i = blockIdx.x * blockDim.x + threadIdx.x;
    if (i >= total) return;
    int e = i >> hshift;
    int hd = i & ((1 << hshift) - 1);
    int s, d;
    if (e < N_EDGES) { s = esrc[e]; d = edst[e]; } else { s = e - N_EDGES; d = s; }
    float v = lrelu_f(asn[(s << hshift) + hd] + adn[(d << hshift) + hd]);
    float* addr = &mx[(d << hshift) + hd];
    if (v >= 0.f) atomicMax((int*)addr, __float_as_int(v));
    else          atomicMin((unsigned int*)addr, __float_as_uint(v));
}

// Edge pass 2: e = exp(v - max[dst]); store e; denom[dst] += e
__global__ void gat_edge_exp(const int* __restrict__ esrc, const int* __restrict__ edst,
                             const float* __restrict__ asn, const float* __restrict__ adn,
                             const float* __restrict__ mx, float* __restrict__ dn,
                             float* __restrict__ ebuf, int hshift, int total)
{
    int i = blockIdx.x * blockDim.x + threadIdx.x;
    if (i >= total) return;
    int e = i >> hshift;
    int hd = i & ((1 << hshift) - 1);
    int s, d;
    if (e < N_EDGES) { s = esrc[e]; d = edst[e]; } else { s = e - N_EDGES; d = s; }
    float v = lrelu_f(asn[(s << hshift) + hd] + adn[(d << hshift) + hd]);
    float ex = __expf(v - mx[(d << hshift) + hd]);
    ebuf[i] = ex;
    unsafeAtomicAdd(&dn[(d << hshift) + hd], ex);
}

// ---------------------------------------------------------------------------
// Edge pass 3: agg[dst, ch..ch+3] += h[src, ch..ch+3] * attn
// One thread per (edge, 4 channels): a wave's 32 lanes cover one edge's 128
// channels with a single b128 gather each; agg (51 MB) stays resident in the
// 192 MB L2 so the f32 atomics are on-chip RMWs.
// ---------------------------------------------------------------------------
__global__ void gat_edge_scatter(const int* __restrict__ esrc, const int* __restrict__ edst,
                                 const float* __restrict__ ebuf, const float* __restrict__ dn,
                                 const float* __restrict__ h, float* __restrict__ agg,
                                 int hshift, int cshift, int total)
{
    int i = blockIdx.x * blockDim.x + threadIdx.x;
    if (i >= total) return;
    int e  = i >> 5;                 // 32 threads per edge
    int ch = (i & 31) << 2;          // 4 channels per thread
    int hd = ch >> cshift;           // head boundaries are >= 16-channel aligned
    int s, d;
    if (e < N_EDGES) { s = esrc[e]; d = edst[e]; } else { s = e - N_EDGES; d = s; }
    float attn = ebuf[(e << hshift) + hd] / (dn[(d << hshift) + hd] + 1e-16f);
    const float4 hv = *(const float4*)(h + ((size_t)s << 7) + ch);
    float* ap = agg + ((size_t)d << 7) + ch;
    unsafeAtomicAdd(ap + 0, hv.x * attn);
    unsafeAtomicAdd(ap + 1, hv.y * attn);
    unsafeAtomicAdd(ap + 2, hv.z * attn);
    unsafeAtomicAdd(ap + 3, hv.w * attn);
}

// ---------------------------------------------------------------------------
// Fused: out = LayerNorm(elu(agg + bias)) * gamma + beta  (+ residual)
// One wave32 per 128-wide row; 4 channels/lane; shfl_xor reductions.
// ---------------------------------------------------------------------------
__global__ __launch_bounds__(256)
void gat_post_ln(const float* __restrict__ agg, const float* __restrict__ bias,
                 const float* __restrict__ gamma, const float* __restrict__ beta,
                 const float* __restrict__ residual, float* __restrict__ outp, int n)
{
    int wave = threadIdx.x >> 5;
    int lane = threadIdx.x & 31;
    int row = blockIdx.x * 8 + wave;
    if (row >= n) return;

    const float4 a4 = ((const float4*)(agg + (size_t)row * HID))[lane];
    const float4 b4 = ((const float4*)bias)[lane];
    float v0 = elu_f(a4.x + b4.x);
    float v1 = elu_f(a4.y + b4.y);
    float v2 = elu_f(a4.z + b4.z);
    float v3 = elu_f(a4.w + b4.w);

    float s  = v0 + v1 + v2 + v3;
    float sq = v0 * v0 + v1 * v1 + v2 * v2 + v3 * v3;
#pragma unroll
    for (int off = 16; off > 0; off >>= 1) {
        s  += __shfl_xor(s,  off, 32);
        sq += __shfl_xor(sq, off, 32);
    }
    float mean = s * (1.f / HID);
    float var  = sq * (1.f / HID) - mean * mean;
    float inv  = rsqrtf(var + 1e-5f);

    const float4 g4  = ((const float4*)gamma)[lane];
    const float4 be4 = ((const float4*)beta)[lane];
    float4 o;
    o.x = (v0 - mean) * inv * g4.x + be4.x;
    o.y = (v1 - mean) * inv * g4.y + be4.y;
    o.z = (v2 - mean) * inv * g4.z + be4.z;
    o.w = (v3 - mean) * inv * g4.w + be4.w;
    if (residual) {
        const float4 r4 = ((const float4*)(residual + (size_t)row * HID))[lane];
        o.x += r4.x; o.y += r4.y; o.z += r4.z; o.w += r4.w;
    }
    ((float4*)(outp + (size_t)row * HID))[lane] = o;
}

// ---------------------------------------------------------------------------
// Global mean pool
// ---------------------------------------------------------------------------
__global__ void pool_acc(const float* __restrict__ h, const int* __restrict__ batch,
                         float* __restrict__ pooled, int total)
{
    int i = blockIdx.x * blockDim.x + threadIdx.x;
    if (i >= total) return;
    int node = i >> 5, ch = (i & 31) << 2;
    const float4 hv = *(const float4*)(h + ((size_t)node << 7) + ch);
    float* pp = pooled + ((size_t)batch[node] << 7) + ch;
    unsafeAtomicAdd(pp + 0, hv.x);
    unsafeAtomicAdd(pp + 1, hv.y);
    unsafeAtomicAdd(pp + 2, hv.z);
    unsafeAtomicAdd(pp + 3, hv.w);
}

__global__ void pool_cnt(const int* __restrict__ batch, float* __restrict__ cnt, int n)
{
    int i = blockIdx.x * blockDim.x + threadIdx.x;
    if (i < n) unsafeAtomicAdd(&cnt[batch[i]], 1.0f);
}

__global__ void pool_div(float* __restrict__ pooled, const float* __restrict__ cnt, int total)
{
    int i = blockIdx.x * blockDim.x + threadIdx.x;
    if (i >= total) return;
    pooled[i] /= fmaxf(cnt[i >> 7], 1.0f);
}

// ---------------------------------------------------------------------------
// Host-side sequencing
// ---------------------------------------------------------------------------
template <int K>
static void run_gat_layer(const float* hin, const float* W, const float* a_s, const float* a_d,
                          const float* b, const float* g, const float* be, const float* res,
                          float* proj, float* agg, float* outp,
                          float* asn, float* adn, float* mx, float* dn, float* ebuf,
                          const int* esrc, const int* edst,
                          int H, int hshift, int cshift, hipStream_t stream)
{
    const int C = HID / H;
    // projection: proj = hin @ W  (WMMA + TDM-staged B)
    wmma_gemm_k<K, 4><<<cdiv(N_NODES, 64), 256, 0, stream>>>(hin, W, proj, nullptr, N_NODES, 0);
    // per-node attention coefficients
    gat_alpha<<<cdiv(N_NODES, 8), 256, 0, stream>>>(proj, a_s, a_d, asn, adn, H, C, N_NODES);
    // init reduction buffers
    fill_f32<<<cdiv((long long)N_NODES * H, 256), 256, 0, stream>>>(mx, -INFINITY, N_NODES * H);
    fill_f32<<<cdiv((long long)N_NODES * H, 256), 256, 0, stream>>>(dn, 0.f, N_NODES * H);
    fill_f32<<<cdiv((long long)N_NODES * HID, 256), 256, 0, stream>>>(agg, 0.f, N_NODES * HID);
    // edge passes
    const int totEH = ETOT * H;
    gat_edge_max<<<cdiv(totEH, 256), 256, 0, stream>>>(esrc, edst, asn, adn, mx, hshift, totEH);
    gat_edge_exp<<<cdiv(totEH, 256), 256, 0, stream>>>(esrc, edst, asn, adn, mx, dn, ebuf,
                                                       hshift, totEH);
    const int totEC = ETOT * 32;     // 4 channels per thread
    gat_edge_scatter<<<cdiv(totEC, 256), 256, 0, stream>>>(esrc, edst, ebuf, dn, proj, agg,
                                                           hshift, cshift, totEC);
    // bias + elu + layernorm (+ residual)
    gat_post_ln<<<cdiv(N_NODES, 8), 256, 0, stream>>>(agg, b, g, be, res, outp, N_NODES);
}

extern "C" void kernel_launch(void* const* d_in, const int* in_sizes, int n_in,
                              void* d_out, int out_size, void* d_ws, size_t ws_size,
                              hipStream_t stream)
{
    const float* x    = (const float*)d_in[0];
    const int*   ei   = (const int*)d_in[1];
    const int*   bat  = (const int*)d_in[2];
    const float* W1   = (const float*)d_in[3];
    const float* as1  = (const float*)d_in[4];
    const float* ad1  = (const float*)d_in[5];
    const float* b1   = (const float*)d_in[6];
    const float* W2   = (const float*)d_in[7];
    const float* as2  = (const float*)d_in[8];
    const float* ad2  = (const float*)d_in[9];
    const float* b2   = (const float*)d_in[10];
    const float* W3   = (const float*)d_in[11];
    const float* as3  = (const float*)d_in[12];
    const float* ad3  = (const float*)d_in[13];
    const float* b3   = (const float*)d_in[14];
    const float* g1   = (const float*)d_in[15];
    const float* be1  = (const float*)d_in[16];
    const float* g2   = (const float*)d_in[17];
    const float* be2  = (const float*)d_in[18];
    const float* g3   = (const float*)d_in[19];
    const float* be3  = (const float*)d_in[20];
    const float* lw1  = (const float*)d_in[21];
    const float* lb1  = (const float*)d_in[22];
    const float* lw2  = (const float*)d_in[23];
    const float* lb2  = (const float*)d_in[24];

    const int* esrc = ei;
    const int* edst = ei + N_EDGES;

    // workspace carve-up (all float)
    float* bufA   = (float*)d_ws;                        // N*HID  (projections / hcur2)
    float* bufB   = bufA + (size_t)N_NODES * HID;        // N*HID  (edge aggregate)
    float* bufC   = bufB + (size_t)N_NODES * HID;        // N*HID  (hcur1 / final feats)
    float* asn    = bufC + (size_t)N_NODES * HID;        // N*HEADS
    float* adn    = asn  + (size_t)N_NODES * HEADS;      // N*HEADS
    float* mx     = adn  + (size_t)N_NODES * HEADS;      // N*HEADS
    float* dn     = mx   + (size_t)N_NODES * HEADS;      // N*HEADS
    float* ebuf   = dn   + (size_t)N_NODES * HEADS;      // ETOT*HEADS
    float* pooled = ebuf + (size_t)ETOT * HEADS;         // G*HID
    float* cnt    = pooled + (size_t)NGRAPH * HID;       // G
    float* t1     = cnt + NGRAPH;                        // G*HID

    // ---- layer 1: x -> bufA(proj) -> bufB(agg) -> bufC (no residual) ----
    run_gat_layer<F_IN>(x, W1, as1, ad1, b1, g1, be1, /*res=*/nullptr,
                        bufA, bufB, bufC, asn, adn, mx, dn, ebuf, esrc, edst,
                        HEADS, /*hshift=*/3, /*cshift=*/4, stream);

    // ---- layer 2: bufC -> bufA(proj) -> bufB(agg) -> bufA (+ bufC residual) ----
    run_gat_layer<HID>(bufC, W2, as2, ad2, b2, g2, be2, /*res=*/bufC,
                       bufA, bufB, bufA, asn, adn, mx, dn, ebuf, esrc, edst,
                       1, /*hshift=*/0, /*cshift=*/7, stream);

    // ---- layer 3: bufA -> bufC(proj) -> bufB(agg) -> bufC (+ bufA residual) ----
    run_gat_layer<HID>(bufA, W3, as3, ad3, b3, g3, be3, /*res=*/bufA,
                       bufC, bufB, bufC, asn, adn, mx, dn, ebuf, esrc, edst,
                       1, /*hshift=*/0, /*cshift=*/7, stream);

    // ---- global mean pool over graphs ----
    fill_f32<<<cdiv(NGRAPH * HID, 256), 256, 0, stream>>>(pooled, 0.f, NGRAPH * HID);
    fill_f32<<<cdiv(NGRAPH, 256), 256, 0, stream>>>(cnt, 0.f, NGRAPH);
    pool_acc<<<cdiv((long long)N_NODES * 32, 256), 256, 0, stream>>>(bufC, bat, pooled,
                                                                     N_NODES * 32);
    pool_cnt<<<cdiv(N_NODES, 256), 256, 0, stream>>>(bat, cnt, N_NODES);
    pool_div<<<cdiv(NGRAPH * HID, 256), 256, 0, stream>>>(pooled, cnt, NGRAPH * HID);

    // ---- heads: elu(pooled@lw1+lb1) @ lw2 + lb2 -> d_out ----
    wmma_gemm_k<HID, 4><<<cdiv(NGRAPH, 64), 256, 0, stream>>>(
        pooled, lw1, t1, lb1, NGRAPH, /*elu=*/1);
    wmma_gemm_k<HID, 4><<<cdiv(NGRAPH, 64), 256, 0, stream>>>(
        t1, lw2, (float*)d_out, lb2, NGRAPH, /*elu=*/0);
}